// QuantumLayer_84645215469641
// MI455X (gfx1250) — compile-verified
//
#include <hip/hip_runtime.h>

// MI455X (gfx1250) — memory-bound quantum-layer kernel.
//
// Math reduction: RX conjugates Z to cos(t)Z + sin(t)Y and <Y>==0 on a real
// state, so   out[b,i] = cos(theta_i) * (sum_k sign(i,k) x[b,k]^2)
//                        / (sum_k x[b,k]^2).
// Roofline: 128 MiB in + 32 MiB out @ 23.3 TB/s => ~7.2 us floor. Everything
// is built around streaming x once with b128 loads.
//
// WMMA tile: D(16x16) = A(16x4) x B(4x16) + C over 4 K-slices
// (V_WMMA_F32_16X16X4_F32, pure f32 — exact vs a VALU implementation).
// K-slot mapping (ours to choose; A is synthesized in-register to match):
//     k = 8*h + 2*j + vgpr      (h = lane>>4, j = K-slice index, vgpr in {0,1})
// so each lane's 8 needed elements are x[sample][8h .. 8h+7]: two contiguous
// float4s -> two global_load_b128 per tile per lane (zero redundant bytes).
//   A rows: M=0..3 Pauli-Z signs, M=4 all-ones (gives ||x||^2), M>4 zero.
//   D rows 0..4 land in acc[0..4] of lanes 0..15 -> one b128 store per sample.
// Fast paths: v_cos_f32 (__cosf) for the 4 uniform weights, v_rcp_f32 for the
// per-sample normalization (q_weights ~ N(0,1); error << any sane tolerance).

typedef __attribute__((ext_vector_type(2))) float v2f;
typedef __attribute__((ext_vector_type(4))) float v4f;
typedef __attribute__((ext_vector_type(8))) float v8f;

#define DIM 16

__device__ __forceinline__ v8f wmma_chain(const v2f a[4], v4f p0, v4f p1) {
  v8f acc = {};
  acc = __builtin_amdgcn_wmma_f32_16x16x4_f32(false, a[0], false, (v2f){p0.x, p0.y},
                                              (short)0, acc, false, false);
  acc = __builtin_amdgcn_wmma_f32_16x16x4_f32(false, a[1], false, (v2f){p0.z, p0.w},
                                              (short)0, acc, false, false);
  acc = __builtin_amdgcn_wmma_f32_16x16x4_f32(false, a[2], false, (v2f){p1.x, p1.y},
                                              (short)0, acc, false, false);
  acc = __builtin_amdgcn_wmma_f32_16x16x4_f32(false, a[3], false, (v2f){p1.z, p1.w},
                                              (short)0, acc, false, false);
  return acc;
}

__global__ __launch_bounds__(256) void qlayer_z_expect_kernel(
    const float* __restrict__ x,     // [B,16]
    const float* __restrict__ qw,    // [4]
    float* __restrict__ out,         // [B,4]
    int ntiles)                      // B/16
{
  const int lane = threadIdx.x & 31;
  const int n    = lane & 15;   // N column (sample in tile) == M row for A
  const int h    = lane >> 4;   // half-wave selector

  const int wave   = (blockIdx.x * blockDim.x + threadIdx.x) >> 5;
  const int nwaves = (gridDim.x * blockDim.x) >> 5;

  // cos(theta_i): uniform, hardware v_cos_f32 (tiny, accurate for N(0,1) args).
  const float cw0 = __cosf(qw[0]);
  const float cw1 = __cosf(qw[1]);
  const float cw2 = __cosf(qw[2]);
  const float cw3 = __cosf(qw[3]);

  // A operand (constant per lane): A[M=n][slot(j,vgpr)] = sign(n, 8h+2j+vgpr).
  // sign(i,k) = +1 if bit (3-i) of k is 0 else -1 (MSB-first wire order).
  v2f a[4];
#pragma unroll
  for (int j = 0; j < 4; ++j) {
    const int k0 = 8 * h + 2 * j;
    float s0, s1;
    if (n < 4) {
      s0 = (((k0    ) >> (3 - n)) & 1) ? -1.0f : 1.0f;
      s1 = (((k0 + 1) >> (3 - n)) & 1) ? -1.0f : 1.0f;
    } else if (n == 4) {
      s0 = 1.0f; s1 = 1.0f;          // ones row -> total probability (norm^2)
    } else {
      s0 = 0.0f; s1 = 0.0f;
    }
    a[j].x = s0;
    a[j].y = s1;
  }

  const int npairs = ntiles >> 1;    // two consecutive tiles per wave iteration

  for (int p = wave; p < npairs; p += nwaves) {
    const int t0 = 2 * p;

    // 2 KB contiguous per wave per iteration; 4 b128 loads per lane-pair tile.
    const float* xs0 = x + ((size_t)t0 * 16 + (size_t)n) * DIM + 8 * h;
    const float* xs1 = xs0 + 16 * DIM;
    v4f r0 = *(const v4f*)(xs0);
    v4f r1 = *(const v4f*)(xs0 + 4);
    v4f q0 = *(const v4f*)(xs1);
    v4f q1 = *(const v4f*)(xs1 + 4);

    // Prefetch the next pair this wave will touch (global_prefetch_b8).
    if (p + nwaves < npairs) {
      __builtin_prefetch(x + ((size_t)(2 * (p + nwaves)) * 16 + (size_t)n) * DIM + 8 * h, 0, 1);
    }

    const v8f acc0 = wmma_chain(a, r0 * r0, r1 * r1);
    const v8f acc1 = wmma_chain(a, q0 * q0, q1 * q1);

    // D rows 0..4 live in acc[0..4] of lanes 0..15 (M = vgpr + 8*h).
    if (lane < 16) {
      const float i0 = __builtin_amdgcn_rcpf(acc0[4]);   // 1/||x||^2, v_rcp_f32
      const float i1 = __builtin_amdgcn_rcpf(acc1[4]);
      v4f o0 = { cw0 * acc0[0] * i0, cw1 * acc0[1] * i0,
                 cw2 * acc0[2] * i0, cw3 * acc0[3] * i0 };
      v4f o1 = { cw0 * acc1[0] * i1, cw1 * acc1[1] * i1,
                 cw2 * acc1[2] * i1, cw3 * acc1[3] * i1 };
      *(v4f*)(out + ((size_t)t0 * 16 + (size_t)n) * 4)       = o0;  // b128 store
      *(v4f*)(out + ((size_t)(t0 + 1) * 16 + (size_t)n) * 4) = o1;  // b128 store
    }
  }

  // Tail: at most one leftover tile if ntiles is odd.
  for (int t = (npairs << 1) + wave; t < ntiles; t += nwaves) {
    const float* xs = x + ((size_t)t * 16 + (size_t)n) * DIM + 8 * h;
    v4f r0 = *(const v4f*)(xs);
    v4f r1 = *(const v4f*)(xs + 4);
    const v8f acc = wmma_chain(a, r0 * r0, r1 * r1);
    if (lane < 16) {
      const float i0 = __builtin_amdgcn_rcpf(acc[4]);
      v4f o = { cw0 * acc[0] * i0, cw1 * acc[1] * i0,
                cw2 * acc[2] * i0, cw3 * acc[3] * i0 };
      *(v4f*)(out + ((size_t)t * 16 + (size_t)n) * 4) = o;
    }
  }
}

extern "C" void kernel_launch(void* const* d_in, const int* in_sizes, int n_in,
                              void* d_out, int out_size, void* d_ws, size_t ws_size,
                              hipStream_t stream) {
  (void)n_in; (void)out_size; (void)d_ws; (void)ws_size;

  const float* x  = (const float*)d_in[0];   // [B,16] f32
  const float* qw = (const float*)d_in[1];   // [4]    f32
  float* out      = (float*)d_out;           // [B,4]  f32

  const int B      = in_sizes[0] / DIM;      // 2,097,152
  const int ntiles = B / 16;                 // 131,072

  const int threads = 256;                   // 8 wave32s per workgroup
  int blocks = 2048;                         // 16,384 waves -> 4 pair-iterations/wave
  const int waves_per_block = threads / 32;
  const int max_blocks = ((ntiles >> 1) + waves_per_block - 1) / waves_per_block;
  if (blocks > max_blocks) blocks = max_blocks;
  if (blocks < 1) blocks = 1;

  qlayer_z_expect_kernel<<<blocks, threads, 0, stream>>>(x, qw, out, ntiles);
}